// CII_18236431138962
// MI455X (gfx1250) — compile-verified
//
#include <hip/hip_runtime.h>
#include <hip/hip_bf16.h>

// Sizes from the reference
#define NN 2048
#define SS 16
#define DD 128
#define HH 8

typedef __attribute__((ext_vector_type(2))) float v2f;
typedef __attribute__((ext_vector_type(8))) float v8f;

// ---------------------------------------------------------------------------
// Kernel 1: per-(n,d) column sort along S; emit diff and positional encoding.
// One thread per column (N*D = 262144 threads).
// ---------------------------------------------------------------------------
__global__ void prep_kernel(const float* __restrict__ x,
                            float* __restrict__ diff,
                            float* __restrict__ enc) {
    int t = blockIdx.x * blockDim.x + threadIdx.x;   // 0 .. N*D-1
    int n = t >> 7;          // / 128
    int d = t & 127;
    const float* xp = x + (size_t)n * (SS * DD) + d;

    float v[SS];
    int   id[SS];
#pragma unroll
    for (int s = 0; s < SS; ++s) { v[s] = xp[s * DD]; id[s] = s; }

    // stable insertion sort, descending (matches stable argsort of -x)
#pragma unroll
    for (int i = 1; i < SS; ++i) {
        float key = v[i]; int kid = id[i];
        int j = i - 1;
        while (j >= 0 && v[j] < key) { v[j + 1] = v[j]; id[j + 1] = id[j]; --j; }
        v[j + 1] = key; id[j + 1] = kid;
    }

    float* dp = diff + (size_t)n * (SS * DD) + d;
    float* ep = enc  + (size_t)n * (SS * DD) + d;
    float csum = 0.0f;
#pragma unroll
    for (int s = 0; s < SS; ++s) {
        float nxt = (s < SS - 1) ? v[s + 1] : 0.0f;   // zero-padded tail
        dp[s * DD] = v[s] - nxt;
        csum += (float)(1u << id[s]);                 // exact powers of two
        ep[s * DD] = csum * (1.0f / 65536.0f);        // / 2^S
    }
}

// ---------------------------------------------------------------------------
// Kernel 2: fused  GEMM1 (enc x Wi) -> LN1 -> PReLU -> GEMM2 (cat x Wj)
// grid = (N/32, H); block = 256 threads (8 wave32).
// M=32 register blocking: every B fragment load feeds TWO wmma ops
// (2 loads/wmma instead of 3 -> relieves the VMEM issue bottleneck).
// Dynamic LDS: se tile [32 rows][S*D = 2048] fp32 = 256 KB (CDNA5 320KB WGP).
// All matrix math on V_WMMA_F32_16X16X4_F32 (exact fp32).
// ---------------------------------------------------------------------------
__global__ void fused_kernel(const float* __restrict__ enc,
                             const float* __restrict__ diff,
                             const float* __restrict__ Wi,
                             const float* __restrict__ Wj,
                             const float* __restrict__ ln1_w,
                             const float* __restrict__ ln1_b,
                             const float* __restrict__ prelu1_a,
                             float* __restrict__ out_pre) {
    extern __shared__ float se[];               // [32][2048]
    const int nt   = blockIdx.x;                // n-tile (32 rows), 0..63
    const int h    = blockIdx.y;                // head,   0..7
    const int tid  = threadIdx.x;
    const int wave = tid >> 5;
    const int lane = tid & 31;
    const int row  = lane & 15;                 // M (or N) index within tile
    const int hi   = lane >> 4;                 // lane half selects K pairs

    const size_t rowStride16 = (size_t)16 * SS * DD;   // 16 batch rows in enc/diff

    // ---------------- Phase 1: se[n0..n0+31, s, o] = enc(n,s,:) @ Wi[h,s] ---
    // 16 s-values * 8 o-tiles = 128 tile-jobs over 8 waves; 2 M-tiles each.
    for (int jj = 0; jj < 16; ++jj) {
        int job = wave * 16 + jj;
        int s  = job >> 3;
        int ot = job & 7;
        const float* A0 = enc + (((size_t)(nt * 32 + row)) * SS + s) * DD;
        const float* A1 = A0 + rowStride16;
        const float* Bb = Wi  + (((size_t)(h * SS + s)) * DD) * DD + ot * 16 + row;
        v8f c0 = {}, c1 = {};
        for (int k = 0; k < DD; k += 4) {
            v2f a0, a1, b;
            a0.x = A0[k + 2 * hi];
            a0.y = A0[k + 2 * hi + 1];
            a1.x = A1[k + 2 * hi];
            a1.y = A1[k + 2 * hi + 1];
            b.x  = Bb[(size_t)(k + 2 * hi) * DD];
            b.y  = Bb[(size_t)(k + 2 * hi + 1) * DD];
            c0 = __builtin_amdgcn_wmma_f32_16x16x4_f32(false, a0, false, b,
                                                       (short)0, c0, false, false);
            c1 = __builtin_amdgcn_wmma_f32_16x16x4_f32(false, a1, false, b,
                                                       (short)0, c1, false, false);
        }
        float* outp = se + s * DD + ot * 16 + row;   // column = row (lane&15)
#pragma unroll
        for (int r = 0; r < 8; ++r) {
            int m = r + 8 * hi;                      // C layout: M=r / M=r+8
            outp[(size_t)m * 2048]        = c0[r];
            outp[(size_t)(m + 16) * 2048] = c1[r];
        }
    }
    __syncthreads();

    // ---------------- Phase 2: LayerNorm over (S,D)=2048 per row + PReLU ----
    const float a1s = *prelu1_a;
    for (int rr = 0; rr < 4; ++rr) {
        int m = wave * 4 + rr;                       // 8 waves * 4 rows = 32
        float* rowp = se + (size_t)m * 2048;
        float sum = 0.0f, sq = 0.0f;
        for (int t = lane; t < 2048; t += 32) {
            float vv = rowp[t]; sum += vv; sq += vv * vv;
        }
        for (int off = 16; off; off >>= 1) {
            sum += __shfl_down(sum, off, 32);
            sq  += __shfl_down(sq,  off, 32);
        }
        sum = __shfl(sum, 0, 32);
        sq  = __shfl(sq,  0, 32);
        float mu  = sum * (1.0f / 2048.0f);
        float var = sq  * (1.0f / 2048.0f) - mu * mu;   // population var
        float rs  = rsqrtf(var + 1e-5f);
        for (int t = lane; t < 2048; t += 32) {
            float vv = (rowp[t] - mu) * rs * ln1_w[t] + ln1_b[t];
            rowp[t] = (vv >= 0.0f) ? vv : a1s * vv;
        }
    }
    __syncthreads();

    // ---------------- Phase 3: out[n,h,o] = sum_{s,d} cat * Wj -------------
    // one o-tile per wave, 2 M-tiles; K = S*256 = 4096 split diff/se halves.
    {
        int ot = wave;                                // 0..7
        const float* Bb = Wj + ((size_t)(h * SS)) * (2 * DD) * DD + ot * 16 + row;
        v8f c0 = {}, c1 = {};
        for (int s = 0; s < SS; ++s) {
            const float* Bs = Bb + (size_t)s * (2 * DD) * DD;
            // first half of cat: diff[n,s,0:128] (global)
            const float* Ad0 = diff + (((size_t)(nt * 32 + row)) * SS + s) * DD;
            const float* Ad1 = Ad0 + rowStride16;
            for (int d0 = 0; d0 < DD; d0 += 4) {
                v2f a0, a1, b;
                a0.x = Ad0[d0 + 2 * hi];
                a0.y = Ad0[d0 + 2 * hi + 1];
                a1.x = Ad1[d0 + 2 * hi];
                a1.y = Ad1[d0 + 2 * hi + 1];
                b.x  = Bs[(size_t)(d0 + 2 * hi) * DD];
                b.y  = Bs[(size_t)(d0 + 2 * hi + 1) * DD];
                c0 = __builtin_amdgcn_wmma_f32_16x16x4_f32(false, a0, false, b,
                                                           (short)0, c0, false, false);
                c1 = __builtin_amdgcn_wmma_f32_16x16x4_f32(false, a1, false, b,
                                                           (short)0, c1, false, false);
            }
            // second half of cat: se[n, s, 0:128] (LDS)
            const float* As0 = se + (size_t)row * 2048 + s * DD;
            const float* As1 = As0 + (size_t)16 * 2048;
            const float* Bs2 = Bs + (size_t)DD * DD;
            for (int d0 = 0; d0 < DD; d0 += 4) {
                v2f a0, a1, b;
                a0.x = As0[d0 + 2 * hi];
                a0.y = As0[d0 + 2 * hi + 1];
                a1.x = As1[d0 + 2 * hi];
                a1.y = As1[d0 + 2 * hi + 1];
                b.x  = Bs2[(size_t)(d0 + 2 * hi) * DD];
                b.y  = Bs2[(size_t)(d0 + 2 * hi + 1) * DD];
                c0 = __builtin_amdgcn_wmma_f32_16x16x4_f32(false, a0, false, b,
                                                           (short)0, c0, false, false);
                c1 = __builtin_amdgcn_wmma_f32_16x16x4_f32(false, a1, false, b,
                                                           (short)0, c1, false, false);
            }
        }
#pragma unroll
        for (int r = 0; r < 8; ++r) {
            int m  = r + 8 * hi;
            int n0 = nt * 32 + m;
            int n1 = n0 + 16;
            out_pre[((size_t)n0 * HH + h) * DD + ot * 16 + row] = c0[r];
            out_pre[((size_t)n1 * HH + h) * DD + ot * 16 + row] = c1[r];
        }
    }
}

// ---------------------------------------------------------------------------
// Kernel 3: in-place LayerNorm over (H,D)=1024 per n + PReLU on d_out.
// grid = N blocks, 256 threads; each thread owns 4 elements.
// ---------------------------------------------------------------------------
__global__ void ln2_kernel(float* __restrict__ out,
                           const float* __restrict__ ln2_w,
                           const float* __restrict__ ln2_b,
                           const float* __restrict__ prelu2_a) {
    __shared__ float ssum[8], ssq[8];
    int n    = blockIdx.x;
    int tid  = threadIdx.x;
    int lane = tid & 31, wave = tid >> 5;
    float* p = out + (size_t)n * (HH * DD);

    float v[4];
    float sum = 0.0f, sq = 0.0f;
#pragma unroll
    for (int i = 0; i < 4; ++i) {
        v[i] = p[tid + 256 * i];
        sum += v[i]; sq += v[i] * v[i];
    }
    for (int off = 16; off; off >>= 1) {
        sum += __shfl_down(sum, off, 32);
        sq  += __shfl_down(sq,  off, 32);
    }
    if (lane == 0) { ssum[wave] = sum; ssq[wave] = sq; }
    __syncthreads();
    if (tid == 0) {
        float S = 0.0f, Q = 0.0f;
        for (int i = 0; i < 8; ++i) { S += ssum[i]; Q += ssq[i]; }
        ssum[0] = S; ssq[0] = Q;
    }
    __syncthreads();
    float mu  = ssum[0] * (1.0f / 1024.0f);
    float var = ssq[0]  * (1.0f / 1024.0f) - mu * mu;
    float rs  = rsqrtf(var + 1e-5f);
    float a2  = *prelu2_a;
#pragma unroll
    for (int i = 0; i < 4; ++i) {
        int j = tid + 256 * i;
        float vv = (v[i] - mu) * rs * ln2_w[j] + ln2_b[j];
        p[j] = (vv >= 0.0f) ? vv : a2 * vv;
    }
}

// ---------------------------------------------------------------------------
extern "C" void kernel_launch(void* const* d_in, const int* in_sizes, int n_in,
                              void* d_out, int out_size, void* d_ws, size_t ws_size,
                              hipStream_t stream) {
    const float* x     = (const float*)d_in[0];
    const float* Wi    = (const float*)d_in[1];
    const float* Wj    = (const float*)d_in[2];
    const float* ln1_w = (const float*)d_in[3];
    const float* ln1_b = (const float*)d_in[4];
    const float* pa1   = (const float*)d_in[5];
    const float* ln2_w = (const float*)d_in[6];
    const float* ln2_b = (const float*)d_in[7];
    const float* pa2   = (const float*)d_in[8];
    float* out = (float*)d_out;

    // workspace: diff [N,S,D] then enc [N,S,D]  (2 x 16 MB)
    float* diff = (float*)d_ws;
    float* enc  = diff + (size_t)NN * SS * DD;

    const int total = NN * DD;                   // 262144 columns
    prep_kernel<<<total / 256, 256, 0, stream>>>(x, diff, enc);

    dim3 grid(NN / 32, HH);                      // 64 x 8 workgroups
    size_t lds_bytes = (size_t)32 * 2048 * sizeof(float);   // 256 KB se tile
    fused_kernel<<<grid, 256, lds_bytes, stream>>>(enc, diff, Wi, Wj,
                                                   ln1_w, ln1_b, pa1, out);

    ln2_kernel<<<NN, 256, 0, stream>>>(out, ln2_w, ln2_b, pa2);
}